// mLSTM_49873160241436
// MI455X (gfx1250) — compile-verified
//
#include <hip/hip_runtime.h>
#include <math.h>

// mLSTM cell for MI455X (gfx1250, wave32).
// B=64, I=1024, H=M=1024.
//
// Cost model: streaming c_prev (256MB) + writing C (256MB) ~= 22us at 23.3TB/s.
// Everything else (6 small GEMMs ~0.8 GFLOP, vectors) is fused/negligible.
// GEMMs use V_WMMA_F32_16X16X4_F32 (fp32 in/out, matches reference precision).
// C-update and C@q are fused so C is touched exactly once; NT hints keep the
// 512MB stream out of the 192MB L2.

typedef __attribute__((ext_vector_type(2))) float v2f;
typedef __attribute__((ext_vector_type(4))) float v4f;
typedef __attribute__((ext_vector_type(8))) float v8f;

#define BATCH 64
#define DIM   1024   // I == H == M

__device__ __forceinline__ float sigmoidf_(float x) {
    return 1.0f / (1.0f + __expf(-x));
}

// ---------------------------------------------------------------------------
// Kernel 1: Y = act(x @ W^T + b) for 6 weight matrices.
// One wave (32 threads) per 16x16 output tile.
// grid = (DIM/16 = 64 N-tiles, BATCH/16 = 4 M-tiles, 6 matrices)
//
// WMMA f32 16x16x4 fragment layout (ISA 7.12.2):
//   A (16x4 MxK): lane%16 = M; lanes<16 hold K=0,1 in v0,v1; lanes>=16 hold K=2,3
//   B (4x16 KxN): lane%16 = N; lanes<16 hold K=0,1;           lanes>=16 hold K=2,3
//   C/D 16x16:    VGPR j: lanes<16 -> M=j, N=lane; lanes>=16 -> M=j+8, N=lane-16
// ---------------------------------------------------------------------------
__global__ __launch_bounds__(32) void gemm6_wmma_kernel(
    const float* __restrict__ x,
    const float* __restrict__ Wq, const float* __restrict__ bq,
    const float* __restrict__ Wk, const float* __restrict__ bk,
    const float* __restrict__ Wv, const float* __restrict__ bv,
    const float* __restrict__ Wi, const float* __restrict__ bi,
    const float* __restrict__ Wf, const float* __restrict__ bf,
    const float* __restrict__ Wo, const float* __restrict__ bo,
    float* __restrict__ qt, float* __restrict__ kt, float* __restrict__ vt,
    float* __restrict__ it, float* __restrict__ ft, float* __restrict__ ot)
{
    const int nt    = blockIdx.x;          // N tile (output column / 16)
    const int mt    = blockIdx.y;          // M tile (batch row / 16)
    const int which = blockIdx.z;          // 0..5 : q,k,v,i,f,o

    const float* W;  const float* bias;  float* out;  int mode;
    switch (which) {
        case 0:  W = Wq; bias = bq; out = qt; mode = 0; break;  // identity
        case 1:  W = Wk; bias = bk; out = kt; mode = 1; break;  // * 1/sqrt(1024)
        case 2:  W = Wv; bias = bv; out = vt; mode = 0; break;  // identity
        case 3:  W = Wi; bias = bi; out = it; mode = 2; break;  // sigmoid
        case 4:  W = Wf; bias = bf; out = ft; mode = 2; break;  // sigmoid
        default: W = Wo; bias = bo; out = ot; mode = 2; break;  // sigmoid
    }

    const int lane = threadIdx.x;          // 0..31
    const int half = lane >> 4;            // 0 | 1 (K sub-pair select)
    const int l16  = lane & 15;

    // A fragment source: x[mt*16 + l16, k + half*2 .. +1]
    const float* ap = x + (size_t)(mt * 16 + l16) * DIM + half * 2;
    // B fragment source: B[k,n] = W^T[k,n] = W[n,k] -> W[nt*16+l16, k+half*2 ..+1]
    const float* bp = W + (size_t)(nt * 16 + l16) * DIM + half * 2;

    v8f acc = {0.f, 0.f, 0.f, 0.f, 0.f, 0.f, 0.f, 0.f};

#pragma unroll 8
    for (int k = 0; k < DIM; k += 4) {
        v2f a = *(const v2f*)(ap + k);
        v2f b = *(const v2f*)(bp + k);
        // (neg_a, A, neg_b, B, c_mod, C, reuse_a, reuse_b)
        acc = __builtin_amdgcn_wmma_f32_16x16x4_f32(
            false, a, false, b, (short)0, acc, false, false);
    }

    const int   col = nt * 16 + l16;
    const float bb  = bias[col];

#pragma unroll
    for (int v = 0; v < 8; ++v) {
        float y = acc[v] + bb;
        if (mode == 1)      y *= 0.03125f;       // 1/sqrt(1024)
        else if (mode == 2) y = sigmoidf_(y);
        acc[v] = y;
    }

    const int row0 = mt * 16 + half * 8;
#pragma unroll
    for (int v = 0; v < 8; ++v)
        out[(size_t)(row0 + v) * DIM + col] = acc[v];
}

// ---------------------------------------------------------------------------
// Kernel 2: n = f*n_prev + i*k ;  denom[b] = max(|n . q|, 1)
// One block (256 threads) per batch; each thread owns 4 contiguous m.
// ---------------------------------------------------------------------------
__global__ __launch_bounds__(256) void n_denom_kernel(
    const float* __restrict__ n_prev,
    const float* __restrict__ qt, const float* __restrict__ kt,
    const float* __restrict__ it, const float* __restrict__ ft,
    float* __restrict__ n_out, float* __restrict__ denom)
{
    const int b   = blockIdx.x;
    const int tid = threadIdx.x;
    const size_t base = (size_t)b * DIM + tid * 4;

    v4f f4 = *(const v4f*)(ft     + base);
    v4f i4 = *(const v4f*)(it     + base);
    v4f k4 = *(const v4f*)(kt     + base);
    v4f p4 = *(const v4f*)(n_prev + base);
    v4f q4 = *(const v4f*)(qt     + base);

    v4f n4 = f4 * p4 + i4 * k4;
    *(v4f*)(n_out + base) = n4;

    float acc = n4[0]*q4[0] + n4[1]*q4[1] + n4[2]*q4[2] + n4[3]*q4[3];

#pragma unroll
    for (int off = 16; off > 0; off >>= 1)
        acc += __shfl_xor(acc, off, 32);

    __shared__ float red[8];
    if ((tid & 31) == 0) red[tid >> 5] = acc;
    __syncthreads();
    if (tid == 0) {
        float s = 0.f;
#pragma unroll
        for (int w = 0; w < 8; ++w) s += red[w];
        denom[b] = fmaxf(fabsf(s), 1.0f);
    }
}

// ---------------------------------------------------------------------------
// Kernel 3 (the 512MB stream): per row m of batch b
//   C[b,m,:] = f*c_prev[b,m,:] + (i*v)*k[b,:]        (NT store, single touch)
//   ht[b,m]  = o * (C[b,m,:] . q[b,:]) / denom[b]    (fused, no C re-read)
// 8 waves per block, one row per wave; kt/qt staged in LDS once per block.
// grid = BATCH * (DIM/8) = 8192 blocks.
// ---------------------------------------------------------------------------
__global__ __launch_bounds__(256) void fuse_C_kernel(
    const float* __restrict__ c_prev,
    const float* __restrict__ qt, const float* __restrict__ kt,
    const float* __restrict__ vt, const float* __restrict__ it,
    const float* __restrict__ ft, const float* __restrict__ ot,
    const float* __restrict__ denom,
    float* __restrict__ C_out, float* __restrict__ ht)
{
    const int b  = blockIdx.x >> 7;          // /128 row-groups per batch
    const int rg = blockIdx.x & 127;
    const int tid  = threadIdx.x;
    const int wave = tid >> 5;
    const int lane = tid & 31;
    const int m    = rg * 8 + wave;

    __shared__ float skt[DIM];
    __shared__ float sqt[DIM];
    {
        const size_t base = (size_t)b * DIM + tid * 4;
        *(v4f*)(&skt[tid * 4]) = *(const v4f*)(kt + base);
        *(v4f*)(&sqt[tid * 4]) = *(const v4f*)(qt + base);
    }
    __syncthreads();

    const size_t vbase = (size_t)b * DIM + m;
    const float f  = ft[vbase];
    const float iv = it[vbase] * vt[vbase];

    const size_t rbase = ((size_t)b * DIM + m) * DIM;
    const float* cp = c_prev + rbase;
    float*       Cp = C_out  + rbase;

    float acc = 0.f;
#pragma unroll
    for (int chunk = 0; chunk < 8; ++chunk) {
        const int k = chunk * 128 + lane * 4;
        v4f c4 = __builtin_nontemporal_load((const v4f*)(cp + k));
        v4f k4 = *(const v4f*)(&skt[k]);
        v4f q4 = *(const v4f*)(&sqt[k]);
        v4f C4 = f * c4 + iv * k4;
        __builtin_nontemporal_store(C4, (v4f*)(Cp + k));
        acc += C4[0]*q4[0] + C4[1]*q4[1] + C4[2]*q4[2] + C4[3]*q4[3];
    }

#pragma unroll
    for (int off = 16; off > 0; off >>= 1)
        acc += __shfl_xor(acc, off, 32);

    if (lane == 0)
        ht[vbase] = ot[vbase] * acc / denom[b];
}

// ---------------------------------------------------------------------------
// Host launcher
// ---------------------------------------------------------------------------
extern "C" void kernel_launch(void* const* d_in, const int* in_sizes, int n_in,
                              void* d_out, int out_size, void* d_ws, size_t ws_size,
                              hipStream_t stream) {
    (void)in_sizes; (void)n_in; (void)out_size; (void)ws_size;

    const float* x      = (const float*)d_in[0];
    const float* c_prev = (const float*)d_in[1];
    const float* n_prev = (const float*)d_in[2];
    const float* Wq = (const float*)d_in[3];  const float* bq = (const float*)d_in[4];
    const float* Wk = (const float*)d_in[5];  const float* bk = (const float*)d_in[6];
    const float* Wv = (const float*)d_in[7];  const float* bv = (const float*)d_in[8];
    const float* Wi = (const float*)d_in[9];  const float* bi = (const float*)d_in[10];
    const float* Wf = (const float*)d_in[11]; const float* bf = (const float*)d_in[12];
    const float* Wo = (const float*)d_in[13]; const float* bo = (const float*)d_in[14];

    // Output tuple (ht, C, n) flattened.
    float* out_ht = (float*)d_out;                                      // 64*1024
    float* out_C  = out_ht + (size_t)BATCH * DIM;                       // 64*1024*1024
    float* out_n  = out_C  + (size_t)BATCH * DIM * DIM;                 // 64*1024

    // Workspace: 6 activation matrices [64,1024] + denom[64].
    float* ws  = (float*)d_ws;
    const size_t SZ = (size_t)BATCH * DIM;
    float* qt = ws;        float* kt = ws + 1*SZ;  float* vt = ws + 2*SZ;
    float* it = ws + 3*SZ; float* ft = ws + 4*SZ;  float* ot = ws + 5*SZ;
    float* denom = ws + 6*SZ;

    // 1) six GEMMs + bias + activation via WMMA
    gemm6_wmma_kernel<<<dim3(DIM / 16, BATCH / 16, 6), 32, 0, stream>>>(
        x, Wq, bq, Wk, bk, Wv, bv, Wi, bi, Wf, bf, Wo, bo,
        qt, kt, vt, it, ft, ot);

    // 2) n and denom
    n_denom_kernel<<<dim3(BATCH), 256, 0, stream>>>(
        n_prev, qt, kt, it, ft, out_n, denom);

    // 3) fused C update + C@q + ht
    fuse_C_kernel<<<dim3(BATCH * (DIM / 8)), 256, 0, stream>>>(
        c_prev, qt, kt, vt, it, ft, ot, denom, out_C, out_ht);
}